// SkeletonModel32_13245679141273
// MI455X (gfx1250) — compile-verified
//
#include <hip/hip_runtime.h>
#include <hip/hip_bf16.h>

// ---------------------------------------------------------------------------
// Batched forward kinematics, H36M 32-joint tree.
//   angles  [B,32,3,3] f32   (1152 B / element, lane-contiguous stream)
//   offsets [32,3]     f32   (384 B shared -> staged into LDS via CDNA5
//                             global_load_async_to_lds, ASYNCcnt-tracked)
//   out     [B,32,3]   f32
// Memory-bound: ~403 MB @ 23.3 TB/s => ~17 us floor; FLOPs are negligible
// (~1.5 FLOP/byte), so scalar VALU FMA is the right compute path on CDNA5.
// ---------------------------------------------------------------------------

#define AS1 __attribute__((address_space(1)))
#define AS3 __attribute__((address_space(3)))

// b128 payload type expected by the async-to-LDS builtin (per hipcc diag):
typedef int v4i __attribute__((__vector_size__(4 * sizeof(int))));

#ifndef __has_builtin
#define __has_builtin(x) 0
#endif

#if __has_builtin(__builtin_amdgcn_global_load_async_to_lds_b128)
#define HAVE_ASYNC_LDS 1
#else
#define HAVE_ASYNC_LDS 0
#warning "gfx1250 async-to-LDS builtin not found; falling back to plain LDS copy"
#endif

#if __has_builtin(__builtin_amdgcn_s_wait_asynccnt)
#define WAIT_ASYNCCNT0() __builtin_amdgcn_s_wait_asynccnt(0)
#else
#define WAIT_ASYNCCNT0() asm volatile("s_wait_asynccnt 0" ::: "memory")
#endif

static constexpr int kNJ = 32;
static constexpr int kPar[kNJ] = {
    -1, 0, 1, 2, 3, 4, 0, 6, 7, 8, 9, 0, 11, 12, 13, 14,
    12, 16, 17, 18, 19, 20, 19, 22, 12, 24, 25, 26, 27, 28, 27, 30};
// joint j needs its global rotation iff it is some joint's parent
static constexpr bool kIsParent[kNJ] = {
    true,  true,  true,  true,  true,  false, true,  true,
    true,  true,  false, true,  true,  true,  true,  false,
    true,  true,  true,  true,  true,  false, true,  false,
    true,  true,  true,  true,  true,  false, true,  false};

__global__ __launch_bounds__(256) void fk32_kernel(
    const float* __restrict__ angles, const float* __restrict__ offsets,
    float* __restrict__ out, int batch) {
  __shared__ float s_off[kNJ * 3];  // 384 B
  const int t = threadIdx.x;

#if HAVE_ASYNC_LDS
  // CDNA5 async global->LDS DMA of the offsets table: 24 lanes x b128 = 96 f32.
  if (t < 24) {
    __builtin_amdgcn_global_load_async_to_lds_b128(
        (AS1 v4i*)(offsets + t * 4), (AS3 v4i*)(&s_off[t * 4]),
        /*offset=*/0, /*cpol=*/0);
  }
  WAIT_ASYNCCNT0();
#else
  if (t < kNJ * 3) s_off[t] = offsets[t];
#endif
  __syncthreads();

  const long long b = (long long)blockIdx.x * blockDim.x + t;
  if (b >= batch) return;

  const float* __restrict__ ab = angles + (size_t)b * (kNJ * 9);
  float* __restrict__ ob = out + (size_t)b * (kNJ * 3);

  // Warm the stream (gfx1250 global_prefetch_b8).
  __builtin_prefetch(ab, 0, 0);
  __builtin_prefetch(ab + 64, 0, 0);

  float R[kNJ][9];  // fully scalarized after unroll; ~3 matrices live max
  float P[kNJ][3];

  // Root: rot[0] = angles[:,0]; pos[0] = 0.
#pragma unroll
  for (int i = 0; i < 9; ++i) R[0][i] = ab[i];
  P[0][0] = 0.f; P[0][1] = 0.f; P[0][2] = 0.f;
  ob[0] = 0.f; ob[1] = 0.f; ob[2] = 0.f;

#pragma unroll
  for (int j = 1; j < kNJ; ++j) {
    const int p = kPar[j];

    // pos[j][l] = sum_k off[j][k] * R[p][k][l] + pos[p][l]
    const float o0 = s_off[3 * j + 0];
    const float o1 = s_off[3 * j + 1];
    const float o2 = s_off[3 * j + 2];
#pragma unroll
    for (int l = 0; l < 3; ++l) {
      const float v =
          fmaf(o0, R[p][0 + l], fmaf(o1, R[p][3 + l], fmaf(o2, R[p][6 + l], P[p][l])));
      P[j][l] = v;
      ob[3 * j + l] = v;
    }

    // rot[j] = angles[j] @ rot[p]  (skip for leaves: never consumed)
    if (kIsParent[j]) {
      const float* __restrict__ aj = ab + 9 * j;
      const float a0 = aj[0], a1 = aj[1], a2 = aj[2];
      const float a3 = aj[3], a4 = aj[4], a5 = aj[5];
      const float a6 = aj[6], a7 = aj[7], a8 = aj[8];
#pragma unroll
      for (int l = 0; l < 3; ++l) {
        R[j][0 + l] = fmaf(a0, R[p][0 + l], fmaf(a1, R[p][3 + l], a2 * R[p][6 + l]));
        R[j][3 + l] = fmaf(a3, R[p][0 + l], fmaf(a4, R[p][3 + l], a5 * R[p][6 + l]));
        R[j][6 + l] = fmaf(a6, R[p][0 + l], fmaf(a7, R[p][3 + l], a8 * R[p][6 + l]));
      }
    }
  }
}

extern "C" void kernel_launch(void* const* d_in, const int* in_sizes, int n_in,
                              void* d_out, int out_size, void* d_ws, size_t ws_size,
                              hipStream_t stream) {
  const float* angles = (const float*)d_in[0];    // [B,32,3,3]
  const float* offsets = (const float*)d_in[1];   // [32,3]
  float* out = (float*)d_out;                     // [B,32,3]

  const int B = in_sizes[0] / (kNJ * 9);
  const int threads = 256;  // 8 wave32 waves / block
  const int blocks = (B + threads - 1) / threads;
  fk32_kernel<<<blocks, threads, 0, stream>>>(angles, offsets, out, B);
}